// AGNN_22574348108380
// MI455X (gfx1250) — compile-verified
//
#include <hip/hip_runtime.h>
#include <hip/hip_bf16.h>

typedef __attribute__((ext_vector_type(2))) float v2f;
typedef __attribute__((ext_vector_type(8))) float v8f;

#define NEG_SLOPE 0.2f

// ---------------------------------------------------------------------------
// GEMM: h = x @ W   (x: [n,64] row-major, W: [64,64] row-major, fp32)
// One wave computes a 16-row x 64-col block using V_WMMA_F32_16X16X4_F32.
// A 16x4 layout: lane L (half=L/16, m=L&15): VGPR v = A[m][kb + 2*half + v]
// B  4x16 layout: lane L: VGPR v = B[kb + 2*half + v][n=L&15]
// C/D 16x16: lane L: VGPR v = C[v + 8*half][n=L&15]
//
// LDS holds W pair-interleaved: sW[(k>>1)*160 + n*2 + (k&1)] = W[k][n], so a
// lane's B register pair (W[kcol][n], W[kcol+1][n]) is one ds_load_b64.
// Pitch 160 (mod 64 dwords = 32) staggers the two half-wave k-pair rows onto
// disjoint bank halves -> conflict-free wave32 b64 loads.
// ---------------------------------------------------------------------------
__global__ void gat_gemm64(const float* __restrict__ x, const float* __restrict__ W,
                           float* __restrict__ h, int nrows) {
  __shared__ float sW[32 * 160];
  int t = threadIdx.x;                       // 256 threads = 8 waves
  for (int i = t; i < 1024; i += 256) {      // 4096 floats as float4
    float4 w4 = ((const float4*)W)[i];       // W[k][n..n+3], k=i>>4, n=(i&15)*4
    int k = i >> 4, n = (i & 15) * 4;
    float* dst = &sW[(k >> 1) * 160 + n * 2 + (k & 1)];
    dst[0] = w4.x; dst[2] = w4.y; dst[4] = w4.z; dst[6] = w4.w;
  }
  __syncthreads();

  int wave = t >> 5;
  int lane = t & 31;
  int half = lane >> 4;
  int l16  = lane & 15;
  long rowBase = ((long)blockIdx.x * 8 + wave) * 16;
  if (rowBase >= nrows) return;              // grid sized exactly; whole-wave uniform

  const float* xrow = x + (rowBase + l16) * 64;
  v8f acc0 = {}, acc1 = {}, acc2 = {}, acc3 = {};

#pragma unroll
  for (int kb = 0; kb < 64; kb += 4) {
    int kcol = kb + 2 * half;                // this half-wave's k pair base
    v2f a = *(const v2f*)&xrow[kcol];        // global_load_b64
    const v2f* brow = (const v2f*)&sW[((kb >> 1) + half) * 160];
    v2f b0 = brow[l16];                      // ds_load_b64, aligned pair
    v2f b1 = brow[16 + l16];
    v2f b2 = brow[32 + l16];
    v2f b3 = brow[48 + l16];
    acc0 = __builtin_amdgcn_wmma_f32_16x16x4_f32(false, a, false, b0, (short)0, acc0, false, false);
    acc1 = __builtin_amdgcn_wmma_f32_16x16x4_f32(false, a, false, b1, (short)0, acc1, false, false);
    acc2 = __builtin_amdgcn_wmma_f32_16x16x4_f32(false, a, false, b2, (short)0, acc2, false, false);
    acc3 = __builtin_amdgcn_wmma_f32_16x16x4_f32(false, a, false, b3, (short)0, acc3, false, false);
  }

#pragma unroll
  for (int v = 0; v < 8; v++) {
    long r = rowBase + v + 8 * half;
    float* hr = h + r * 64 + l16;
    hr[0]  = acc0[v];
    hr[16] = acc1[v];
    hr[32] = acc2[v];
    hr[48] = acc3[v];
  }
}

// ---------------------------------------------------------------------------
// Per-node attention logits: als[i] = h[i,:]·a_src,  ald[i] = h[i,:]·a_dst
// ---------------------------------------------------------------------------
__global__ void gat_logits(const float* __restrict__ h, const float* __restrict__ a_src,
                           const float* __restrict__ a_dst, float* __restrict__ als,
                           float* __restrict__ ald, int n) {
  long i = (long)blockIdx.x * blockDim.x + threadIdx.x;
  if (i >= n) return;
  const float4* hp = (const float4*)(h + i * 64);
  float s = 0.f, d = 0.f;
#pragma unroll
  for (int j = 0; j < 16; j++) {
    float4 v  = hp[j];
    float4 as = ((const float4*)a_src)[j];
    float4 ad = ((const float4*)a_dst)[j];
    s += v.x * as.x + v.y * as.y + v.z * as.z + v.w * as.w;
    d += v.x * ad.x + v.y * ad.y + v.z * ad.z + v.w * ad.w;
  }
  als[i] = s; ald[i] = d;
}

// ---------------------------------------------------------------------------
// Init per-layer accumulators: m = -inf, denom = 0, out = 0
// ---------------------------------------------------------------------------
__global__ void gat_init(float* __restrict__ m, float* __restrict__ den,
                         float* __restrict__ o, int n) {
  long i = (long)blockIdx.x * blockDim.x + threadIdx.x;
  if (i < n) { m[i] = -__builtin_inff(); den[i] = 0.f; }
  if (i < (long)n * 64) o[i] = 0.f;
}

__device__ __forceinline__ void edge_sd(const int* __restrict__ ei, int E, long e,
                                        int& s, int& d) {
  if (e < E) { s = ei[e]; d = ei[(long)E + e]; }
  else       { s = d = (int)(e - E); }          // self loop
}

__device__ __forceinline__ void atomicMaxF(float* addr, float val) {
  if (val >= 0.f) atomicMax((int*)addr, __float_as_int(val));
  else            atomicMin((unsigned int*)addr, __float_as_uint(val));
}

// Pass 1: segment max of leaky_relu(als[src]+ald[dst]) over dst
__global__ void gat_edge_max(const int* __restrict__ ei, int E, int n,
                             const float* __restrict__ als, const float* __restrict__ ald,
                             float* __restrict__ m) {
  long e = (long)blockIdx.x * blockDim.x + threadIdx.x;
  long Et = (long)E + n;
  if (e >= Et) return;
  int s, d; edge_sd(ei, E, e, s, d);
  float v = als[s] + ald[d];
  v = v > 0.f ? v : NEG_SLOPE * v;
  atomicMaxF(&m[d], v);
}

// Pass 2: ex = exp(e - m[dst]); stash ex; denom[dst] += ex
__global__ void gat_edge_exp(const int* __restrict__ ei, int E, int n,
                             const float* __restrict__ als, const float* __restrict__ ald,
                             const float* __restrict__ m, float* __restrict__ exb,
                             float* __restrict__ den) {
  long e = (long)blockIdx.x * blockDim.x + threadIdx.x;
  long Et = (long)E + n;
  if (e >= Et) return;
  int s, d; edge_sd(ei, E, e, s, d);
  float v = als[s] + ald[d];
  v = v > 0.f ? v : NEG_SLOPE * v;
  float ex = __expf(v - m[d]);
  exb[e] = ex;
  atomicAdd(&den[d], ex);
}

// Pass 3: out[dst,:] += alpha * h[src,:]   (16 threads/edge, float4 chunks)
__global__ void gat_edge_scatter(const int* __restrict__ ei, int E, int n,
                                 const float* __restrict__ exb, const float* __restrict__ den,
                                 const float* __restrict__ h, float* __restrict__ o) {
  long g = (long)blockIdx.x * blockDim.x + threadIdx.x;
  long Et = (long)E + n;
  if (g >= Et * 16) return;
  long e = g >> 4;
  int  c = (int)(g & 15);
  int s, d; edge_sd(ei, E, e, s, d);
  float alpha = exb[e] / (den[d] + 1e-16f);
  float4 hv = *(const float4*)(h + (long)s * 64 + c * 4);
  float* op = o + (long)d * 64 + c * 4;
  atomicAdd(op + 0, hv.x * alpha);
  atomicAdd(op + 1, hv.y * alpha);
  atomicAdd(op + 2, hv.z * alpha);
  atomicAdd(op + 3, hv.w * alpha);
}

// out = relu(out + b)
__global__ void gat_bias_relu(float* __restrict__ o, const float* __restrict__ b, long total) {
  long i = (long)blockIdx.x * blockDim.x + threadIdx.x;
  if (i >= total) return;
  float v = o[i] + b[i & 63];
  o[i] = v > 0.f ? v : 0.f;
}

// Final: out[g] = h[g, 0:1280]·W_out + b_out   (one wave per row)
__global__ void gat_readout(const float* __restrict__ h, const float* __restrict__ Wout,
                            const float* __restrict__ bout, float* __restrict__ out, int ng) {
  long wv = ((long)blockIdx.x * blockDim.x + threadIdx.x) >> 5;
  int lane = threadIdx.x & 31;
  if (wv >= ng) return;
  const float* row = h + wv * 1280;
  float s = 0.f;
#pragma unroll
  for (int j0 = 0; j0 < 1280; j0 += 128) {
    int j = j0 + lane * 4;
    float4 v = *(const float4*)(row + j);
    float4 w = *(const float4*)(Wout + j);
    s += v.x * w.x + v.y * w.y + v.z * w.z + v.w * w.w;
  }
  for (int off = 16; off > 0; off >>= 1) s += __shfl_down(s, off, 32);
  if (lane == 0) out[wv] = s + bout[0];
}

// ---------------------------------------------------------------------------
extern "C" void kernel_launch(void* const* d_in, const int* in_sizes, int n_in,
                              void* d_out, int out_size, void* d_ws, size_t ws_size,
                              hipStream_t stream) {
  const float* x    = (const float*)d_in[0];
  const int*   ei   = (const int*)  d_in[1];
  const float* W1   = (const float*)d_in[2];
  const float* as1  = (const float*)d_in[3];
  const float* ad1  = (const float*)d_in[4];
  const float* b1   = (const float*)d_in[5];
  const float* W2   = (const float*)d_in[6];
  const float* as2  = (const float*)d_in[7];
  const float* ad2  = (const float*)d_in[8];
  const float* b2   = (const float*)d_in[9];
  const float* Wout = (const float*)d_in[10];
  const float* bout = (const float*)d_in[11];

  const int N = in_sizes[0] / 64;            // 163840
  const int E = in_sizes[1] / 2;             // 2621440
  const long Et = (long)E + N;

  float* h   = (float*)d_ws;                 // [N,64]
  float* o   = h   + (long)N * 64;           // [N,64]
  float* als = o   + (long)N * 64;           // [N]
  float* ald = als + N;                      // [N]
  float* m   = ald + N;                      // [N]
  float* den = m   + N;                      // [N]
  float* exb = den + N;                      // [Et]

  const int T = 256;
  int gGemm  = N / 128;                      // 8 waves x 16 rows per block
  int gNode  = (N + T - 1) / T;
  long gFeat = ((long)N * 64 + T - 1) / T;
  long gEdge = (Et + T - 1) / T;
  long gScat = (Et * 16 + T - 1) / T;

  // ---------------- Layer 1 (input = x) ----------------
  gat_gemm64      <<<gGemm, T, 0, stream>>>(x, W1, h, N);
  gat_logits      <<<gNode, T, 0, stream>>>(h, as1, ad1, als, ald, N);
  gat_init        <<<gFeat, T, 0, stream>>>(m, den, o, N);
  gat_edge_max    <<<gEdge, T, 0, stream>>>(ei, E, N, als, ald, m);
  gat_edge_exp    <<<gEdge, T, 0, stream>>>(ei, E, N, als, ald, m, exb, den);
  gat_edge_scatter<<<gScat, T, 0, stream>>>(ei, E, N, exb, den, h, o);
  gat_bias_relu   <<<gFeat, T, 0, stream>>>(o, b1, (long)N * 64);

  // ---------------- Layer 2 (input = o, reuse h; then re-zero o) -------
  gat_gemm64      <<<gGemm, T, 0, stream>>>(o, W2, h, N);
  gat_logits      <<<gNode, T, 0, stream>>>(h, as2, ad2, als, ald, N);
  gat_init        <<<gFeat, T, 0, stream>>>(m, den, o, N);
  gat_edge_max    <<<gEdge, T, 0, stream>>>(ei, E, N, als, ald, m);
  gat_edge_exp    <<<gEdge, T, 0, stream>>>(ei, E, N, als, ald, m, exb, den);
  gat_edge_scatter<<<gScat, T, 0, stream>>>(ei, E, N, exb, den, h, o);
  gat_bias_relu   <<<gFeat, T, 0, stream>>>(o, b2, (long)N * 64);

  // ---------------- Readout: [N/20, 1280] @ [1280,1] + b ----------------
  int ng = out_size;                         // 8192 rows
  long gRead = ((long)ng * 32 + T - 1) / T;
  gat_readout     <<<gRead, T, 0, stream>>>(o, Wout, bout, (float*)d_out, ng);
}